// S4_10797547782206
// MI455X (gfx1250) — compile-verified
//
#include <hip/hip_runtime.h>
#include <hip/hip_bf16.h>

#define BATCH 2
#define CCH   512
#define TT    2048
#define HH    8
#define DD    64
#define RR    2
#define GG    8
#define EPSF  1e-5f

typedef float v2f __attribute__((ext_vector_type(2)));
typedef float v8f __attribute__((ext_vector_type(8)));

// ---------------------------------------------------------------------------
// K1: LayerNorm over channel axis per (b,t)
// ---------------------------------------------------------------------------
__global__ __launch_bounds__(256) void k_layernorm(
    const float* __restrict__ x, const float* __restrict__ ln_w,
    const float* __restrict__ ln_b, float* __restrict__ xln) {
  int idx = blockIdx.x * blockDim.x + threadIdx.x;   // over B*T
  int b = idx / TT, t = idx % TT;
  const float* xp = x + (size_t)b * CCH * TT + t;
  float s = 0.f, sq = 0.f;
  for (int c = 0; c < CCH; ++c) {
    float v = xp[(size_t)c * TT];
    s += v; sq += v * v;
  }
  float mu  = s * (1.f / CCH);
  float var = sq * (1.f / CCH) - mu * mu;
  float inv = rsqrtf(var + EPSF);
  float* op = xln + (size_t)b * CCH * TT + t;
  for (int c = 0; c < CCH; ++c) {
    float v = xp[(size_t)c * TT];
    op[(size_t)c * TT] = (v - mu) * inv * ln_w[c] + ln_b[c];
  }
}

// ---------------------------------------------------------------------------
// K2: structured long conv via 64 exponential scans per channel.
//   S_dd[t] = sum_u e^{u*A_dd} * xpad[t-1023+u]   (stable descending recur.)
//   y1[b,c,t] = sum_dd Cfull[h,dd,e] * S_dd[t] + pos_bias[h,t]
// One block per (b,c); 64 lanes = one decay component each.
// The channel row is staged into LDS with GLOBAL_LOAD_ASYNC_TO_LDS_B128
// (ASYNCcnt path, no VGPR round-trip); the copy overlaps the per-lane
// exp/Cfull setup below, then s_wait_asynccnt 0 + barrier publish it.
// ---------------------------------------------------------------------------
__global__ __launch_bounds__(64) void k_scan_conv(
    const float* __restrict__ xln, const float* __restrict__ C1,
    const float* __restrict__ C2, const float* __restrict__ C0,
    const float* __restrict__ pos_bias, float* __restrict__ y1) {
  __shared__ __align__(16) float xrow[TT];
  __shared__ float prod[64 * 17];   // [dd][i], stride 17 avoids bank conflicts

  int tid = threadIdx.x;            // dd = tid
  int ch  = blockIdx.x % CCH;
  int b   = blockIdx.x / CCH;
  int h   = ch / DD;
  int e   = ch % DD;

  const float* xp = xln + ((size_t)b * CCH + ch) * TT;

  // Async memory->LDS stage of the 8KB channel row (8 x B128 per lane).
  unsigned xbase = (unsigned)(uintptr_t)(void*)xrow;   // LDS byte offset
  for (int i = tid; i < TT / 4; i += 64) {
    const float* gaddr = xp + (size_t)i * 4;
    unsigned lds_off = xbase + (unsigned)i * 16u;
    asm volatile("global_load_async_to_lds_b128 %0, %1, off"
                 :: "v"(lds_off), "v"(gaddr) : "memory");
  }

  int dd = tid;
  float A     = -(float)(dd + 1) * (1.f / DD);
  float eA    = expf(A);
  float eTail = expf(2048.f * A);                       // may underflow to 0
  // Cfull[h,dd,e] = C1[h,dd,:] . C2[h,:,e] + (dd==e)*C0[h,e]
  float cf = C1[((size_t)h * DD + dd) * RR + 0] * C2[((size_t)h * RR + 0) * DD + e]
           + C1[((size_t)h * DD + dd) * RR + 1] * C2[((size_t)h * RR + 1) * DD + e]
           + ((dd == e) ? C0[(size_t)h * DD + e] : 0.f);

  asm volatile("s_wait_asynccnt 0x0" ::: "memory");
  __syncthreads();

  // warm-up: t in [T, T+1023], only the leading-edge term is in range
  float S = 0.f;
  for (int t = TT + 1023; t >= TT; --t) S = xrow[t - 1023] + eA * S;

  float* yrow = y1 + ((size_t)b * CCH + ch) * TT;
  const float* pb = pos_bias + (size_t)h * TT;
  for (int chunk = (TT / 16) - 1; chunk >= 0; --chunk) {
    for (int i = 15; i >= 0; --i) {
      int t = chunk * 16 + i;
      float xm = (t >= 1023) ? xrow[t - 1023] : 0.f;
      float xr = (t <= 1022) ? xrow[t + 1025] : 0.f;
      S = xm + eA * S - eTail * xr;
      prod[dd * 17 + i] = cf * S;
    }
    __syncthreads();
    if (tid < 16) {
      int t = chunk * 16 + tid;
      float acc = 0.f;
      for (int j = 0; j < 64; ++j) acc += prod[j * 17 + tid];
      yrow[t] = acc + pb[t];
    }
    __syncthreads();
  }
}

// ---------------------------------------------------------------------------
// K3: z[b,r,t] = sum_c Vf[c,r] * xln[b,c,t]   (rank-2 projection)
// ---------------------------------------------------------------------------
__global__ __launch_bounds__(256) void k_zproj(
    const float* __restrict__ xln, const float* __restrict__ Vf,
    float* __restrict__ z) {
  int idx = blockIdx.x * blockDim.x + threadIdx.x;   // over B*T
  int b = idx / TT, t = idx % TT;
  const float* xp = xln + (size_t)b * CCH * TT + t;
  float z0 = 0.f, z1 = 0.f;
  for (int c = 0; c < CCH; ++c) {
    float v = xp[(size_t)c * TT];
    z0 += Vf[c * RR + 0] * v;
    z1 += Vf[c * RR + 1] * v;
  }
  z[(size_t)b * RR * TT + t]      = z0;
  z[(size_t)b * RR * TT + TT + t] = z1;
}

// ---------------------------------------------------------------------------
// K4: grouped local conv: 3-tap temporal, 64x64 channel mixing per head
// ---------------------------------------------------------------------------
__global__ __launch_bounds__(256) void k_local_conv(
    const float* __restrict__ y1, const float* __restrict__ local_w,
    const float* __restrict__ local_b, float* __restrict__ y2) {
  int tid = threadIdx.x;
  int blk = blockIdx.x;                       // B*C*(T/256)
  int tch = blk % (TT / 256);
  int co  = (blk / (TT / 256)) % CCH;
  int b   = blk / ((TT / 256) * CCH);
  int t   = tch * 256 + tid;
  int h   = co / DD;

  const float* wrow = local_w + (size_t)co * DD * 3;
  const float* yin  = y1 + ((size_t)b * CCH + h * DD) * TT;
  float acc = local_b[co];
  for (int i = 0; i < DD; ++i) {
    const float* row = yin + (size_t)i * TT;
    float x0 = (t >= 1)      ? row[t - 1] : 0.f;
    float x1 =                 row[t];
    float x2 = (t + 1 < TT)  ? row[t + 1] : 0.f;
    acc += wrow[i * 3 + 0] * x0 + wrow[i * 3 + 1] * x1 + wrow[i * 3 + 2] * x2;
  }
  y2[((size_t)b * CCH + co) * TT + t] = acc;
}

// ---------------------------------------------------------------------------
// K5: gate projection with V_WMMA_F32_16X16X4_F32.
// One wave per 16x16 (e,t) tile; computes both gate halves (rows e and 512+e)
// sharing the B fragment, then fuses bias + a*silu(b) + rank-2 low-rank add.
// A frag (16x4 f32): lane l -> row l%16, K = (l/16)*2 + v
// B frag (4x16 f32): lane l -> col l%16, K = (l/16)*2 + v
// C/D:               lane l, vgpr v -> M = v + 8*(l/16), N = l%16
// ---------------------------------------------------------------------------
__global__ __launch_bounds__(256) void k_gate_wmma(
    const float* __restrict__ y2, const float* __restrict__ gate_w,
    const float* __restrict__ gate_b, const float* __restrict__ Uf,
    const float* __restrict__ z, float* __restrict__ out) {
  int lane = threadIdx.x & 31;
  int wid  = threadIdx.x >> 5;
  int gw   = blockIdx.x * 8 + wid;            // global wave-tile id
  int tIdx = gw & 127;                        // T/16 = 128 tiles
  int eIdx = (gw >> 7) & 31;                  // C/16 = 32 tiles
  int b    = gw >> 12;
  int t0 = tIdx * 16, e0 = eIdx * 16;
  int m16 = lane & 15, kk = lane >> 4;

  const float* wa   = gate_w + (size_t)(e0 + m16) * CCH;
  const float* wb   = gate_w + (size_t)(CCH + e0 + m16) * CCH;
  const float* ycol = y2 + (size_t)b * CCH * TT + (t0 + m16);

  __builtin_prefetch(wa, 0, 0);
  __builtin_prefetch(wb, 0, 0);

  v8f accA = {0.f, 0.f, 0.f, 0.f, 0.f, 0.f, 0.f, 0.f};
  v8f accB = {0.f, 0.f, 0.f, 0.f, 0.f, 0.f, 0.f, 0.f};

  for (int k0 = 0; k0 < CCH; k0 += 4) {
    int ka = k0 + kk * 2;
    v2f Aa; Aa.x = wa[ka];   Aa.y = wa[ka + 1];
    v2f Ab; Ab.x = wb[ka];   Ab.y = wb[ka + 1];
    v2f Bf; Bf.x = ycol[(size_t)ka * TT]; Bf.y = ycol[(size_t)(ka + 1) * TT];
    accA = __builtin_amdgcn_wmma_f32_16x16x4_f32(
        false, Aa, false, Bf, (short)0, accA, false, false);
    accB = __builtin_amdgcn_wmma_f32_16x16x4_f32(
        false, Ab, false, Bf, (short)0, accB, false, false);
  }

  int t = t0 + m16;
  float z0 = z[(size_t)b * RR * TT + t];
  float z1 = z[(size_t)b * RR * TT + TT + t];
  float* op = out + (size_t)b * CCH * TT + t;
#pragma unroll
  for (int v = 0; v < 8; ++v) {
    int e = e0 + v + 8 * kk;
    float ga = accA[v] + gate_b[e];
    float gb = accB[v] + gate_b[CCH + e];
    float sil = gb / (1.f + expf(-gb));
    float val = ga * sil + Uf[e * RR + 0] * z0 + Uf[e * RR + 1] * z1;
    op[(size_t)e * TT] = val;
  }
}

// ---------------------------------------------------------------------------
// K6: GroupNorm statistics per (b,g) over contiguous 64*T region
// ---------------------------------------------------------------------------
__global__ __launch_bounds__(256) void k_gn_stats(
    const float* __restrict__ y, float* __restrict__ stats) {
  __shared__ float ssum[256];
  __shared__ float ssq[256];
  int bg = blockIdx.x;                    // B*G
  int b = bg / GG, g = bg % GG;
  const int N = (CCH / GG) * TT;          // 131072
  const float* base = y + (size_t)b * CCH * TT + (size_t)g * (CCH / GG) * TT;
  float s = 0.f, sq = 0.f;
  for (int i = threadIdx.x; i < N; i += 256) {
    float v = base[i];
    s += v; sq += v * v;
  }
  ssum[threadIdx.x] = s; ssq[threadIdx.x] = sq;
  __syncthreads();
  for (int off = 128; off > 0; off >>= 1) {
    if (threadIdx.x < off) {
      ssum[threadIdx.x] += ssum[threadIdx.x + off];
      ssq[threadIdx.x]  += ssq[threadIdx.x + off];
    }
    __syncthreads();
  }
  if (threadIdx.x == 0) {
    float mu  = ssum[0] / (float)N;
    float var = ssq[0] / (float)N - mu * mu;
    stats[bg * 2 + 0] = mu;
    stats[bg * 2 + 1] = var;
  }
}

// ---------------------------------------------------------------------------
// K7: GroupNorm apply (in place on d_out)
// ---------------------------------------------------------------------------
__global__ __launch_bounds__(256) void k_gn_apply(
    float* __restrict__ y, const float* __restrict__ stats,
    const float* __restrict__ gn_w, const float* __restrict__ gn_b) {
  int idx = blockIdx.x * blockDim.x + threadIdx.x;   // over B*C*T
  int b = idx / (CCH * TT);
  int rem = idx % (CCH * TT);
  int c = rem / TT;
  int g = c / (CCH / GG);
  float mu  = stats[(b * GG + g) * 2 + 0];
  float var = stats[(b * GG + g) * 2 + 1];
  float inv = rsqrtf(var + EPSF);
  y[idx] = (y[idx] - mu) * inv * gn_w[c] + gn_b[c];
}

// ---------------------------------------------------------------------------
extern "C" void kernel_launch(void* const* d_in, const int* in_sizes, int n_in,
                              void* d_out, int out_size, void* d_ws, size_t ws_size,
                              hipStream_t stream) {
  (void)in_sizes; (void)n_in; (void)out_size; (void)ws_size;
  const float* x        = (const float*)d_in[0];
  const float* ln_w     = (const float*)d_in[1];
  const float* ln_b     = (const float*)d_in[2];
  const float* C1       = (const float*)d_in[3];
  const float* C2       = (const float*)d_in[4];
  const float* C0       = (const float*)d_in[5];
  const float* pos_bias = (const float*)d_in[6];
  const float* local_w  = (const float*)d_in[7];
  const float* local_b  = (const float*)d_in[8];
  const float* gate_w   = (const float*)d_in[9];
  const float* gate_b   = (const float*)d_in[10];
  const float* U        = (const float*)d_in[11];
  const float* V        = (const float*)d_in[12];
  const float* gn_w     = (const float*)d_in[13];
  const float* gn_b     = (const float*)d_in[14];
  float* out = (float*)d_out;

  const size_t nBCT = (size_t)BATCH * CCH * TT;
  float* xln   = (float*)d_ws;
  float* y1    = xln + nBCT;
  float* y2    = y1 + nBCT;
  float* zbuf  = y2 + nBCT;
  float* stats = zbuf + (size_t)BATCH * RR * TT;

  k_layernorm<<<(BATCH * TT) / 256, 256, 0, stream>>>(x, ln_w, ln_b, xln);
  k_scan_conv<<<BATCH * CCH, 64, 0, stream>>>(xln, C1, C2, C0, pos_bias, y1);
  k_zproj<<<(BATCH * TT) / 256, 256, 0, stream>>>(xln, V, zbuf);
  k_local_conv<<<BATCH * CCH * (TT / 256), 256, 0, stream>>>(y1, local_w, local_b, y2);
  k_gate_wmma<<<(BATCH * (CCH / 16) * (TT / 16)) / 8, 256, 0, stream>>>(
      y2, gate_w, gate_b, U, zbuf, out);
  k_gn_stats<<<BATCH * GG, 256, 0, stream>>>(out, stats);
  k_gn_apply<<<(int)(nBCT / 256), 256, 0, stream>>>(out, stats, gn_w, gn_b);
}